// ParameterKernel_4140348473904
// MI455X (gfx1250) — compile-verified
//
#include <hip/hip_runtime.h>
#include <cstddef>

// CDNA5 / gfx1250, wave32. Fused masked-kernel KRR prediction:
//   y[p,j] = sum_i [Zref[i]==Z[j]] * Alpha[p,i] * (desc_ref[i] . desc[j])^expK
// Two chained GEMMs fused through registers/LDS; matrix math on
// v_wmma_f32_16x16x4_f32 (f32 in / f32 accumulate, matching reference dtype).
// LDS operands are staged in WMMA-lane-packed layout with padded strides so
// every ds_load_b128 is bank-conflict-minimal and feeds 2 k-steps per load.

typedef __attribute__((ext_vector_type(2))) float v2f;
typedef __attribute__((ext_vector_type(8))) float v8f;

namespace {
constexpr int D     = 256;          // d_feat
constexpr int JB    = 64;           // output columns (input atoms) per workgroup
constexpr int IB    = 16;           // ref atoms per i-iteration
constexpr int NW    = 4;            // waves per workgroup
constexpr int BLOCK = NW * 32;
constexpr int KS    = D / 4;        // 64 WMMA k-steps over the descriptor dim
constexpr int PSTR  = 2 * KS + 4;   // 132: packed-lane stride -> banks 4*lane mod 64
constexpr int KTSTR = 20;           // kT row stride -> conflict-free b64 reads
}

__device__ __forceinline__ v8f wmma_f32(v2f a, v2f b, v8f c) {
    return __builtin_amdgcn_wmma_f32_16x16x4_f32(
        false, a, false, b, (short)0, c, false, false);
}

__global__ __launch_bounds__(BLOCK) void krr_fused_wmma(
    const float* __restrict__ Alpha,   // [P=64, N_ref]
    const int*   __restrict__ Zr,      // [N_ref]
    const float* __restrict__ Xref,    // [N_ref, D]
    const int*   __restrict__ Zi,      // [N_in]
    const float* __restrict__ Xin,     // [N_in, D]
    const int*   __restrict__ expk_p,  // [1]
    float*       __restrict__ y,       // [64, N_in]
    int N_ref, int N_in)
{
    // Packed WMMA operand layout: p[lane][2*kstep + s] holds the two floats
    // lane needs for k-step kstep (lanes 0-15: dims {4k,4k+1}; 16-31: {4k+2,4k+3}).
    __shared__ float pB[NW][32][PSTR];     // X_in block, packed per wave  (~66 KB)
    __shared__ float pA[32][PSTR];         // X_ref chunk, packed, shared  (~17 KB)
    __shared__ float kT[NW][16][KTSTR];    // masked K tile, transposed [n][m]
    __shared__ int   zr_s[IB];

    const int tid  = threadIdx.x;
    const int w    = tid >> 5;
    const int lane = tid & 31;
    const int half = lane >> 4;
    const int nrow = lane & 15;
    const int j0   = blockIdx.x * JB;
    const int expk = expk_p[0];

    // ---- stage X_in j-block into packed-B layout (once) ----
    for (int idx = tid; idx < JB * KS; idx += BLOCK) {
        const int row = idx >> 6;             // 0..63   (KS == 64)
        const int kb  = idx & (KS - 1);       // k-step
        const float4 g = ((const float4*)(Xin + (size_t)(j0 + row) * D))[kb];
        const int wv = row >> 4, r16 = row & 15;
        *(v2f*)&pB[wv][r16][2 * kb]      = (v2f){g.x, g.y};   // half 0 dims
        *(v2f*)&pB[wv][16 + r16][2 * kb] = (v2f){g.z, g.w};   // half 1 dims
    }
    const int zin = Zi[j0 + w * 16 + nrow];   // element of this lane's column

    v8f acc[4];
    #pragma unroll
    for (int pt = 0; pt < 4; ++pt) acc[pt] = (v8f){};

    __syncthreads();

    for (int i0 = 0; i0 < N_ref; i0 += IB) {
        // ---- stage X_ref chunk into packed-A layout + Z_ref chunk ----
        for (int idx = tid; idx < IB * KS; idx += BLOCK) {
            const int row = idx >> 6, kb = idx & (KS - 1);
            const float4 g = ((const float4*)(Xref + (size_t)(i0 + row) * D))[kb];
            *(v2f*)&pA[row][2 * kb]      = (v2f){g.x, g.y};
            *(v2f*)&pA[16 + row][2 * kb] = (v2f){g.z, g.w};
        }
        if (tid < IB) zr_s[tid] = Zr[i0 + tid];
        // L2 prefetch of the next chunk (global_prefetch_b8)
        if (i0 + IB < N_ref) {
            const char* nxt = (const char*)(Xref + (size_t)(i0 + IB) * D);
            __builtin_prefetch(nxt + tid * 128, 0, 3);
            __builtin_prefetch(nxt + tid * 128 + 64, 0, 3);
        }
        __syncthreads();

        // ---- GEMM 1: Ktile[16x16] over D; 1 ds_load_b128 per operand feeds
        //      2 WMMAs; dual accumulators break the D-register RAW chain ----
        v8f c0 = (v8f){}, c1 = (v8f){};
        #pragma unroll 8
        for (int kb = 0; kb < KS; kb += 2) {
            const float4 a4 = *(const float4*)&pA[lane][2 * kb];
            const float4 b4 = *(const float4*)&pB[w][lane][2 * kb];
            c0 = wmma_f32((v2f){a4.x, a4.y}, (v2f){b4.x, b4.y}, c0);
            c1 = wmma_f32((v2f){a4.z, a4.w}, (v2f){b4.z, b4.w}, c1);
        }

        // ---- element-match mask + integer power ----
        float kv[8], pw[8], vals[8];
        #pragma unroll
        for (int r = 0; r < 8; ++r) { kv[r] = c0[r] + c1[r]; pw[r] = 1.0f; }
        for (int t = 0; t < expk; ++t) {     // single runtime loop (expK small)
            #pragma unroll
            for (int r = 0; r < 8; ++r) pw[r] *= kv[r];
        }
        #pragma unroll
        for (int r = 0; r < 8; ++r)
            vals[r] = (zr_s[r + 8 * half] == zin) ? pw[r] : 0.0f;

        // spill transposed tile: kT[w][n][m] (same-wave producer/consumer)
        *(float4*)&kT[w][nrow][8 * half]     = make_float4(vals[0], vals[1], vals[2], vals[3]);
        *(float4*)&kT[w][nrow][8 * half + 4] = make_float4(vals[4], vals[5], vals[6], vals[7]);
        __syncthreads();   // all pA/zr_s readers done -> next staging may overwrite

        // ---- GEMM 2: Y[64 x 16] += Alpha[:, i0:i0+16] @ Ktile (4 x K=4) ----
        v2f b2[4];
        #pragma unroll
        for (int kk = 0; kk < 4; ++kk)
            b2[kk] = *(const v2f*)&kT[w][nrow][4 * kk + 2 * half];

        #pragma unroll
        for (int pt = 0; pt < 4; ++pt) {
            const float* arow =
                Alpha + (size_t)(pt * 16 + nrow) * N_ref + i0 + 2 * half;
            #pragma unroll
            for (int kk = 0; kk < 4; ++kk) {
                const v2f a2 = *(const v2f*)(arow + 4 * kk);
                acc[pt] = wmma_f32(a2, b2[kk], acc[pt]);
            }
        }
        // no barrier needed here: kT is wave-private, pA rewrite is fenced above
    }

    // ---- epilogue: each (p, j) written exactly once ----
    const int j = j0 + w * 16 + nrow;
    #pragma unroll
    for (int pt = 0; pt < 4; ++pt) {
        #pragma unroll
        for (int r = 0; r < 8; ++r) {
            const int p = pt * 16 + r + 8 * half;
            y[(size_t)p * N_in + j] = acc[pt][r];
        }
    }
}

extern "C" void kernel_launch(void* const* d_in, const int* in_sizes, int n_in,
                              void* d_out, int out_size, void* d_ws, size_t ws_size,
                              hipStream_t stream) {
    (void)n_in; (void)out_size; (void)d_ws; (void)ws_size;
    const float* Alpha = (const float*)d_in[0];
    const int*   Zr    = (const int*)d_in[1];
    const float* Xref  = (const float*)d_in[2];
    const int*   Zi    = (const int*)d_in[3];
    const float* Xin   = (const float*)d_in[4];
    const int*   expk  = (const int*)d_in[5];
    float*       yout  = (float*)d_out;

    const int N_ref = in_sizes[1];   // 12800
    const int N_in  = in_sizes[3];   // 25600

    const int nblk = N_in / JB;      // 400 workgroups
    hipLaunchKernelGGL(krr_fused_wmma, dim3(nblk), dim3(BLOCK), 0, stream,
                       Alpha, Zr, Xref, Zi, Xin, expk, yout, N_ref, N_in);
}